// DigitCaps_37838661878239
// MI455X (gfx1250) — compile-verified
//
#include <hip/hip_runtime.h>

typedef __attribute__((ext_vector_type(2))) float v2f;
typedef __attribute__((ext_vector_type(4))) float v4f;
typedef __attribute__((ext_vector_type(8))) float v8f;

#define NJ   166
#define NR   864
#define NO   8
#define NI   4
#define NB   128
#define BT   16          // batch tile (= WMMA N)
#define NTHREADS 256
#define NWAVES 8
#define RPW  (NR / NWAVES)   // 108 r-values per wave

// LDS layout (in floats)
#define UT_OFF   0                    // u transposed: [NR][2(khi)][BT][2(klo)] = NR*64
#define BIJ_OFF  (NR * 64)            // routing logits: [BT][NR]
#define S_OFF    (BIJ_OFF + BT * NR)  // s_j: [BT][NO]
#define V_OFF    (S_OFF + BT * NO)    // v_j: [BT][NO]
#define RED_OFF  (V_OFF + BT * NO)    // reduction scratch: [16][16]
#define STM_OFF  (RED_OFF + 256)      // softmax max per b: [16]
#define STZ_OFF  (STM_OFF + 16)       // softmax 1/Z per b: [16]
#define LDS_FLOATS (STZ_OFF + 16)

__global__ __launch_bounds__(NTHREADS, 1)
void digitcaps_routing(const float* __restrict__ u,
                       const float* __restrict__ W,
                       float* __restrict__ out)
{
    extern __shared__ float lds[];
    float* u_t = lds + UT_OFF;
    float* bij = lds + BIJ_OFF;
    float* sj  = lds + S_OFF;
    float* vj  = lds + V_OFF;
    float* red = lds + RED_OFF;
    float* stm = lds + STM_OFF;
    float* stz = lds + STZ_OFF;

    const int j    = blockIdx.x;
    const int b0   = blockIdx.y * BT;
    const int tid  = threadIdx.x;
    const int lane = tid & 31;
    const int wave = tid >> 5;
    const int bl   = lane & 15;     // D: batch column index (lanes 0-15 valid)
    const int khi  = lane >> 4;     // 0 -> i{0,1}, 1 -> i{2,3}
    const bool lo16 = (lane < 16);  // lanes holding D rows 0..7 (the real o's)

    // ---- Stage u tile into LDS in WMMA-B layout; zero routing logits ----
    for (int idx = tid; idx < BT * NR; idx += NTHREADS) {
        const int b = idx / NR;
        const int r = idx - b * NR;
        const v4f u4 = *(const v4f*)(u + (size_t)(b0 + b) * (NR * NI) + r * NI);
        *(v2f*)(u_t + r * 64 +      b * 2) = (v2f){u4.x, u4.y};   // K=0,1
        *(v2f*)(u_t + r * 64 + 32 + b * 2) = (v2f){u4.z, u4.w};   // K=2,3
    }
    for (int idx = tid; idx < BT * NR; idx += NTHREADS) bij[idx] = 0.0f;
    __syncthreads();

    // Branch-free A operand: lanes with (lane&15)>=8 load duplicate W rows (o = bl&7).
    // Their products land in D rows 8..15 == lanes 16..31, which are never read.
    const int rbase = wave * RPW;
    const int aoff  = (bl & 7) * NI + khi * 2;   // within one r: word offset into W[j][r]
    const int boff  = khi * 32 + bl * 2;         // within one r: word offset into u_t row
    const float* __restrict__ wpw = W + (size_t)j * (NR * NO * NI)
                                      + (size_t)rbase * (NO * NI) + aoff;
    const float* __restrict__ utw = u_t + rbase * 64 + boff;
    float* __restrict__ bijl = bij + bl * NR + rbase;   // this lane's logit chunk
    const v8f czero = {0.f,0.f,0.f,0.f,0.f,0.f,0.f,0.f};

    for (int it = 0; it < 3; ++it) {
        if (it != 0) {
            // ---- softmax stats over r, one 16-thread group per batch b ----
            const int gb = tid >> 4;
            const int gt = tid & 15;
            float m = -3.4e38f;
            for (int r = gt; r < NR; r += 16) m = fmaxf(m, bij[gb * NR + r]);
            red[gb * 16 + gt] = m;
            __syncthreads();
            for (int off = 8; off > 0; off >>= 1) {
                if (gt < off) red[gb*16+gt] = fmaxf(red[gb*16+gt], red[gb*16+gt+off]);
                __syncthreads();
            }
            const float mb = red[gb * 16];
            __syncthreads();
            float zs = 0.0f;
            for (int r = gt; r < NR; r += 16) zs += __expf(bij[gb * NR + r] - mb);
            red[gb * 16 + gt] = zs;
            __syncthreads();
            for (int off = 8; off > 0; off >>= 1) {
                if (gt < off) red[gb*16+gt] += red[gb*16+gt+off];
                __syncthreads();
            }
            if (gt == 0) { stm[gb] = mb; stz[gb] = 1.0f / red[gb * 16]; }
        }
        if (tid < BT * NO) sj[tid] = 0.0f;
        __syncthreads();

        // ---- S pass: stream r, recompute u_hat via WMMA, weight by c, accumulate ----
        v8f acc = czero;
        if (it == 0) {
            // b_ij == 0 -> c uniform = 1/NR: no softmax stats, no exp in the loop
            #pragma unroll 4
            for (int rr = 0; rr < RPW; ++rr) {
                const v2f a  = *(const v2f*)(wpw + rr * (NO * NI));
                const v2f bb = *(const v2f*)(utw + rr * 64);
                const v8f d  = __builtin_amdgcn_wmma_f32_16x16x4_f32(
                                   false, a, false, bb, (short)0, czero, false, false);
                acc += d;
            }
            acc *= (1.0f / (float)NR);
        } else {
            const float mbl = stm[bl];
            const float rz  = stz[bl];
            #pragma unroll 4
            for (int rr = 0; rr < RPW; ++rr) {
                const v2f a  = *(const v2f*)(wpw + rr * (NO * NI));
                const v2f bb = *(const v2f*)(utw + rr * 64);
                const v8f d  = __builtin_amdgcn_wmma_f32_16x16x4_f32(
                                   false, a, false, bb, (short)0, czero, false, false);
                const float c = __expf(bijl[rr] - mbl) * rz;
                acc += d * c;
            }
        }
        if (lo16) {
            #pragma unroll
            for (int k = 0; k < NO; ++k) atomicAdd(&sj[bl * NO + k], acc[k]);  // ds_add_f32
        }
        __syncthreads();

        // ---- squash ----
        if (tid < BT) {
            float n2 = 0.0f;
            #pragma unroll
            for (int k = 0; k < NO; ++k) { const float x = sj[tid*NO+k]; n2 += x * x; }
            const float sc = sqrtf(n2) / (1.0f + n2);
            #pragma unroll
            for (int k = 0; k < NO; ++k) vj[tid * NO + k] = sj[tid * NO + k] * sc;
        }
        __syncthreads();

        // ---- A pass: agreement a = <u_hat, v>, update logits (skip on last iter) ----
        if (it < 2) {
            float vk[NO];
            #pragma unroll
            for (int k = 0; k < NO; ++k) vk[k] = vj[bl * NO + k];
            #pragma unroll 4
            for (int rr = 0; rr < RPW; ++rr) {
                const v2f a  = *(const v2f*)(wpw + rr * (NO * NI));
                const v2f bb = *(const v2f*)(utw + rr * 64);
                const v8f d  = __builtin_amdgcn_wmma_f32_16x16x4_f32(
                                   false, a, false, bb, (short)0, czero, false, false);
                float dot = 0.0f;
                #pragma unroll
                for (int k = 0; k < NO; ++k) dot += d[k] * vk[k];
                if (lo16) bijl[rr] += dot;   // unique (b,r) owner per lane
            }
            __syncthreads();
        }
    }

    // ---- write v_j [B, J, O] ----
    if (tid < BT * NO) {
        const int b = tid >> 3;
        const int k = tid & 7;
        out[(size_t)(b0 + b) * (NJ * NO) + j * NO + k] = vj[tid];
    }
}

extern "C" void kernel_launch(void* const* d_in, const int* in_sizes, int n_in,
                              void* d_out, int out_size, void* d_ws, size_t ws_size,
                              hipStream_t stream) {
    const float* u = (const float*)d_in[0];   // [128, 864, 4]
    const float* W = (const float*)d_in[1];   // [1, 166, 864, 8, 4]
    float* out = (float*)d_out;               // [128, 166, 8]
    dim3 grid(NJ, NB / BT);                   // 166 x 8 = 1328 workgroups
    dim3 block(NTHREADS);                     // 8 wave32 waves
    const size_t shmem = (size_t)LDS_FLOATS * sizeof(float);  // ~272 KB < 320 KB/WGP
    hipLaunchKernelGGL(digitcaps_routing, grid, block, shmem, stream, u, W, out);
}